// EdgeSmoothing_58222576665062
// MI455X (gfx1250) — compile-verified
//
#include <hip/hip_runtime.h>
#include <stdint.h>

#define BLOCK 256
#define TILE_EDGES 2048            // edges staged per tile (16 KB per buffer)
#define CHUNKS 4                   // 16B async chunks per thread: 256*4*16B = 16384B = 2048 edges
#define EPT 8                      // edges processed per thread per tile: 2048/256

// Non-returning device-scope float atomic add with immediate byte offset.
// Guarantees native global_atomic_add_f32 (no CAS expansion).
#define GATOMIC_ADD_F32(baseull, OFF, val)                                         \
  asm volatile("global_atomic_add_f32 %0, %1, off offset:" #OFF " scope:SCOPE_DEV" \
               :: "v"(baseull), "v"(val) : "memory")

// ---------------------------------------------------------------------------
// Per-edge work: flow = 0.5*(feat[s]+feat[d]); atomic-add into out[s,8:16]
// and out[d,8:16]. Gathers are 2x b128 per endpoint (L2-resident).
// ---------------------------------------------------------------------------
__device__ __forceinline__ void es_process_edge(int2 e,
                                                const float* __restrict__ nf,
                                                float* __restrict__ out) {
  const float4* ps = (const float4*)nf + ((long long)e.x * 2);
  const float4* pd = (const float4*)nf + ((long long)e.y * 2);
  float4 a0 = ps[0], a1 = ps[1];
  float4 b0 = pd[0], b1 = pd[1];

  float f0 = 0.5f * (a0.x + b0.x);
  float f1 = 0.5f * (a0.y + b0.y);
  float f2 = 0.5f * (a0.z + b0.z);
  float f3 = 0.5f * (a0.w + b0.w);
  float f4 = 0.5f * (a1.x + b1.x);
  float f5 = 0.5f * (a1.y + b1.y);
  float f6 = 0.5f * (a1.z + b1.z);
  float f7 = 0.5f * (a1.w + b1.w);

  unsigned long long rs =
      (unsigned long long)(uintptr_t)(out + (long long)e.x * 16 + 8);
  unsigned long long rd =
      (unsigned long long)(uintptr_t)(out + (long long)e.y * 16 + 8);

  GATOMIC_ADD_F32(rs, 0, f0);
  GATOMIC_ADD_F32(rs, 4, f1);
  GATOMIC_ADD_F32(rs, 8, f2);
  GATOMIC_ADD_F32(rs, 12, f3);
  GATOMIC_ADD_F32(rs, 16, f4);
  GATOMIC_ADD_F32(rs, 20, f5);
  GATOMIC_ADD_F32(rs, 24, f6);
  GATOMIC_ADD_F32(rs, 28, f7);

  GATOMIC_ADD_F32(rd, 0, f0);
  GATOMIC_ADD_F32(rd, 4, f1);
  GATOMIC_ADD_F32(rd, 8, f2);
  GATOMIC_ADD_F32(rd, 12, f3);
  GATOMIC_ADD_F32(rd, 16, f4);
  GATOMIC_ADD_F32(rd, 20, f5);
  GATOMIC_ADD_F32(rd, 24, f6);
  GATOMIC_ADD_F32(rd, 28, f7);
}

// ---------------------------------------------------------------------------
// Main edge kernel: persistent blocks, FULL tiles only (no guards in the hot
// loop). Edge stream double-buffered through LDS with async-to-LDS copies.
// ---------------------------------------------------------------------------
__global__ __launch_bounds__(BLOCK)
void es_edge_scatter_kernel(const int2* __restrict__ edges,
                            const float* __restrict__ nf,
                            float* __restrict__ out,
                            long long full_tiles) {
  __shared__ int2 sh[2][TILE_EDGES];

  const int tid = threadIdx.x;
  long long t = blockIdx.x;
  if (t >= full_tiles) return;

  const unsigned long long gbase = (unsigned long long)(uintptr_t)edges;
  const unsigned lds0 = (unsigned)(uintptr_t)&sh[0][0];
  const unsigned lds1 = (unsigned)(uintptr_t)&sh[1][0];

  // Stage one full 2048-edge tile: exactly CHUNKS async instructions per wave.
  auto stage = [&](long long tt, unsigned ldsbase) {
#pragma unroll
    for (int k = 0; k < CHUNKS; ++k) {
      int chunk = tid + k * BLOCK;
      unsigned long long g = gbase +
          (unsigned long long)tt * (TILE_EDGES * 8ull) +
          (unsigned long long)chunk * 16ull;
      unsigned l = ldsbase + (unsigned)chunk * 16u;
      asm volatile("global_load_async_to_lds_b128 %0, %1, off"
                   :: "v"(l), "v"(g) : "memory");
    }
  };

  stage(t, lds0);  // prologue

  int it = 0;
  for (; t < full_tiles; t += gridDim.x, ++it) {
    const int cur = it & 1;
    long long tn = t + gridDim.x;
    if (tn < full_tiles) {
      stage(tn, cur ? lds0 : lds1);
      asm volatile("s_wait_asynccnt 4" ::: "memory");  // current tile landed
    } else {
      asm volatile("s_wait_asynccnt 0" ::: "memory");
    }
    __syncthreads();

#pragma unroll
    for (int j = 0; j < EPT; ++j) {
      int pos = tid + j * BLOCK;
      int2 e = sh[cur][pos];  // direct shared-array index -> ds_load_b64
      es_process_edge(e, nf, out);
    }
    __syncthreads();  // all waves done reading before this buffer is reused
  }
}

// ---------------------------------------------------------------------------
// Tail kernel: remaining (< TILE_EDGES) edges with direct global loads.
// ---------------------------------------------------------------------------
__global__ void es_edge_tail_kernel(const int2* __restrict__ edges,
                                    const float* __restrict__ nf,
                                    float* __restrict__ out,
                                    long long start, long long n_edges) {
  long long eid = start + (long long)blockIdx.x * blockDim.x + threadIdx.x;
  if (eid >= n_edges) return;
  int2 e = edges[eid];
  es_process_edge(e, nf, out);
}

// ---------------------------------------------------------------------------
// Init: out[n, 0:8] = to_concat[n, :]; out[n, 8:16] = 0   (all b128 traffic)
// ---------------------------------------------------------------------------
__global__ void es_init_kernel(const float4* __restrict__ tc4,
                               float4* __restrict__ out4, int n_nodes) {
  int n = blockIdx.x * blockDim.x + threadIdx.x;
  if (n >= n_nodes) return;
  float4 a = tc4[n * 2 + 0];
  float4 b = tc4[n * 2 + 1];
  float4 z = make_float4(0.0f, 0.0f, 0.0f, 0.0f);
  out4[n * 4 + 0] = a;
  out4[n * 4 + 1] = b;
  out4[n * 4 + 2] = z;
  out4[n * 4 + 3] = z;
}

// ---------------------------------------------------------------------------
// Finalize: out[n, 8:16] /= count[n]   (b128 read-modify-write per node)
// ---------------------------------------------------------------------------
__global__ void es_finalize_kernel(const float* __restrict__ count,
                                   float4* __restrict__ out4, int n_nodes) {
  int n = blockIdx.x * blockDim.x + threadIdx.x;
  if (n >= n_nodes) return;
  float c = count[n];
  float4 x = out4[n * 4 + 2];
  float4 y = out4[n * 4 + 3];
  x.x /= c; x.y /= c; x.z /= c; x.w /= c;
  y.x /= c; y.y /= c; y.z /= c; y.w /= c;
  out4[n * 4 + 2] = x;
  out4[n * 4 + 3] = y;
}

// ---------------------------------------------------------------------------
extern "C" void kernel_launch(void* const* d_in, const int* in_sizes, int n_in,
                              void* d_out, int out_size, void* d_ws, size_t ws_size,
                              hipStream_t stream) {
  const float*  to_concat     = (const float*)d_in[0];
  const float*  node_features = (const float*)d_in[1];
  const int2*   edges         = (const int2*)d_in[2];
  const float*  count         = (const float*)d_in[3];
  float*        out           = (float*)d_out;

  const int       n_nodes = in_sizes[0] / 8;
  const long long n_edges = (long long)in_sizes[2] / 2;

  // 1) init output (copy to_concat, zero accumulator columns)
  {
    int blocks = (n_nodes + BLOCK - 1) / BLOCK;
    es_init_kernel<<<blocks, BLOCK, 0, stream>>>((const float4*)to_concat,
                                                 (float4*)out, n_nodes);
  }

  // 2) edge scatter: full tiles through the async/LDS pipeline ...
  const long long full_tiles = n_edges / TILE_EDGES;
  if (full_tiles > 0) {
    int blocks = (int)((full_tiles < 1024) ? full_tiles : 1024);
    es_edge_scatter_kernel<<<blocks, BLOCK, 0, stream>>>(edges, node_features,
                                                         out, full_tiles);
  }
  // ... and the remainder (< 2048 edges) directly from global
  const long long tail_start = full_tiles * (long long)TILE_EDGES;
  const long long tail = n_edges - tail_start;
  if (tail > 0) {
    int blocks = (int)((tail + BLOCK - 1) / BLOCK);
    es_edge_tail_kernel<<<blocks, BLOCK, 0, stream>>>(edges, node_features, out,
                                                      tail_start, n_edges);
  }

  // 3) divide accumulator by count
  {
    int blocks = (n_nodes + BLOCK - 1) / BLOCK;
    es_finalize_kernel<<<blocks, BLOCK, 0, stream>>>(count, (float4*)out,
                                                     n_nodes);
  }
}